// MetaScSmoother_55645596286990
// MI455X (gfx1250) — compile-verified
//
#include <hip/hip_runtime.h>
#include <hip/hip_bf16.h>

// ---------------------------------------------------------------------------
// MetaScSmoother for MI455X (gfx1250).
// B=64, MU=256, N=65536, ML=3, KS=7, L=7.
//
// Workspace layout (floats):
//   G     [64][7][65536]   ch0 = r, ch1..3 = G1, ch4..6 = G2      (29,360,128)
//   W1    [64][147]        dynamic 7x7 weights, 1->3 conv         (     9,408)
//   W2    [64][441]        dynamic 7x7 weights, 3->3 conv         (    28,224)
//   GRAM  [64][8][8]       P = T T^T, T = [S(7 rows); r]          (     4,096)
//   KV    [64][8]          solved coefficients                    (       512)
// total ~117.6 MB (fits the 192 MB L2).
// ---------------------------------------------------------------------------

#define B    64
#define MU   256
#define NPIX 65536
#define LCH  7

typedef float v2f __attribute__((ext_vector_type(2)));
typedef float v8f __attribute__((ext_vector_type(8)));

// ---------------- meta-MLPs: kA(9) -> 200 -> {147, 441} --------------------
__global__ void mlp_kernel(const float* __restrict__ kA,
                           const float* __restrict__ fc1_w1, const float* __restrict__ fc1_b1,
                           const float* __restrict__ fc1_w2, const float* __restrict__ fc1_b2,
                           const float* __restrict__ fc2_w1, const float* __restrict__ fc2_b1,
                           const float* __restrict__ fc2_w2, const float* __restrict__ fc2_b2,
                           float* __restrict__ W1, float* __restrict__ W2) {
    const int b = blockIdx.x;
    const int t = threadIdx.x;
    __shared__ float sk[9];
    __shared__ float h1[200];
    __shared__ float h2[200];
    if (t < 9) sk[t] = kA[b * 9 + t];
    __syncthreads();
    if (t < 200) {
        float a = fc1_b1[t], c = fc2_b1[t];
        #pragma unroll
        for (int i = 0; i < 9; ++i) {
            a += sk[i] * fc1_w1[i * 200 + t];
            c += sk[i] * fc2_w1[i * 200 + t];
        }
        h1[t] = fmaxf(a, 0.f);
        h2[t] = fmaxf(c, 0.f);
    }
    __syncthreads();
    if (t < 147) {
        float a = fc1_b2[t];
        for (int j = 0; j < 200; ++j) a += h1[j] * fc1_w2[j * 147 + t];
        W1[b * 147 + t] = a;
    }
    for (int o = t; o < 441; o += 256) {
        float a = fc2_b2[o];
        for (int j = 0; j < 200; ++j) a += h2[j] * fc2_w2[j * 441 + o];
        W2[b * 441 + o] = a;
    }
}

// ---------------- r = f - A*x (3x3 stencil, zero pad) ----------------------
__global__ void residual_kernel(const float* __restrict__ x, const float* __restrict__ f,
                                const float* __restrict__ kA, float* __restrict__ G) {
    const int b = blockIdx.y;
    const int p = blockIdx.x * 256 + threadIdx.x;
    __shared__ float sk[9];
    if (threadIdx.x < 9) sk[threadIdx.x] = kA[b * 9 + threadIdx.x];
    __syncthreads();
    const int yy = p >> 8, xx = p & 255;
    const float* xb = x + (size_t)b * NPIX;
    float acc = 0.f;
    #pragma unroll
    for (int dy = -1; dy <= 1; ++dy) {
        int y2 = yy + dy;
        if (y2 < 0 || y2 > 255) continue;
        #pragma unroll
        for (int dx = -1; dx <= 1; ++dx) {
            int x2 = xx + dx;
            if (x2 < 0 || x2 > 255) continue;
            acc += sk[(dy + 1) * 3 + (dx + 1)] * xb[(y2 << 8) + x2];
        }
    }
    G[(size_t)b * LCH * NPIX + p] = f[(size_t)b * NPIX + p] - acc;
}

// ---------------- G1 = conv7x7(r, W1): 1 -> 3 channels ---------------------
#define TS   16
#define HALO 3
#define TD   22
__global__ void conv1_kernel(const float* __restrict__ W1, float* __restrict__ G) {
    const int b  = blockIdx.z;
    const int tx = threadIdx.x & 15, ty = threadIdx.x >> 4;
    const int x0 = blockIdx.x * TS, y0 = blockIdx.y * TS;
    __shared__ float tile[TD][TD];
    __shared__ float w[147];
    const float* r = G + (size_t)b * LCH * NPIX;  // channel 0
    for (int i = threadIdx.x; i < 147; i += 256) w[i] = W1[b * 147 + i];
    for (int i = threadIdx.x; i < TD * TD; i += 256) {
        int ly = i / TD, lx = i % TD;
        int gy = y0 + ly - HALO, gx = x0 + lx - HALO;
        tile[ly][lx] = (gy >= 0 && gy < MU && gx >= 0 && gx < MU) ? r[(gy << 8) + gx] : 0.f;
    }
    __syncthreads();
    float a0 = 0.f, a1 = 0.f, a2 = 0.f;
    #pragma unroll
    for (int ky = 0; ky < 7; ++ky) {
        #pragma unroll
        for (int kx = 0; kx < 7; ++kx) {
            float v = tile[ty + ky][tx + kx];
            int wi = ky * 7 + kx;
            a0 += w[wi]       * v;
            a1 += w[49 + wi]  * v;
            a2 += w[98 + wi]  * v;
        }
    }
    const int p = ((y0 + ty) << 8) + (x0 + tx);
    float* Gb = G + (size_t)b * LCH * NPIX;
    Gb[1 * NPIX + p] = a0;
    Gb[2 * NPIX + p] = a1;
    Gb[3 * NPIX + p] = a2;
}

// ---------------- G2 = conv7x7(G1, W2): 3 -> 3 channels --------------------
__global__ void conv2_kernel(const float* __restrict__ W2, float* __restrict__ G) {
    const int b  = blockIdx.z;
    const int tx = threadIdx.x & 15, ty = threadIdx.x >> 4;
    const int x0 = blockIdx.x * TS, y0 = blockIdx.y * TS;
    __shared__ float tile[3][TD][TD];
    __shared__ float w[441];
    float* Gb = G + (size_t)b * LCH * NPIX;
    for (int i = threadIdx.x; i < 441; i += 256) w[i] = W2[b * 441 + i];
    for (int i = threadIdx.x; i < 3 * TD * TD; i += 256) {
        int c = i / (TD * TD), rest = i % (TD * TD);
        int ly = rest / TD, lx = rest % TD;
        int gy = y0 + ly - HALO, gx = x0 + lx - HALO;
        tile[c][ly][lx] = (gy >= 0 && gy < MU && gx >= 0 && gx < MU)
                              ? Gb[(1 + c) * NPIX + (gy << 8) + gx] : 0.f;
    }
    __syncthreads();
    float acc[3] = {0.f, 0.f, 0.f};
    #pragma unroll
    for (int ci = 0; ci < 3; ++ci) {
        #pragma unroll
        for (int ky = 0; ky < 7; ++ky) {
            #pragma unroll
            for (int kx = 0; kx < 7; ++kx) {
                float v = tile[ci][ty + ky][tx + kx];
                int wi = ci * 49 + ky * 7 + kx;
                acc[0] += w[wi]        * v;   // (o=0,ci)
                acc[1] += w[147 + wi]  * v;   // (o=1,ci)
                acc[2] += w[294 + wi]  * v;   // (o=2,ci)
            }
        }
    }
    const int p = ((y0 + ty) << 8) + (x0 + tx);
    Gb[4 * NPIX + p] = acc[0];
    Gb[5 * NPIX + p] = acc[1];
    Gb[6 * NPIX + p] = acc[2];
}

// ---------------- zero the Gram accumulators -------------------------------
__global__ void zero_kernel(float* __restrict__ g, int n) {
    int i = blockIdx.x * blockDim.x + threadIdx.x;
    if (i < n) g[i] = 0.f;
}

// ---------------- Gram: P = T T^T with V_WMMA_F32_16X16X4_F32 --------------
// T rows 0..6 = S_l = depthwise3x3(G_l); row 7 = r; rows 8..15 = 0.
// P[m][n] (m,n<7) = M; P[m][7] = (S r)_m.  One wave per block; 16 waves/sample,
// each wave reduces 4096 pixels of K; partials combined via f32 global atomics.
__global__ void __launch_bounds__(32)
gram_kernel(const float* __restrict__ kA, const float* __restrict__ G,
            float* __restrict__ gram) {
    const int b    = blockIdx.y;
    const int wid  = blockIdx.x;     // 0..15
    const int lane = threadIdx.x;    // 0..31
    __shared__ float T[16][65];      // pad to 65: (m + c) % 64 distinct banks
    __shared__ float sk[9];
    if (lane < 9) sk[lane] = kA[b * 9 + lane];
    for (int i = lane; i < 16 * 65; i += 32) (&T[0][0])[i] = 0.f;  // rows 8..15 stay zero
    __syncthreads();

    const float* Gb = G + (size_t)b * LCH * NPIX;
    v8f acc = {0.f, 0.f, 0.f, 0.f, 0.f, 0.f, 0.f, 0.f};
    const int m     = lane & 15;
    const int khalf = (lane >> 4) * 2;

    const int pbeg = wid * 4096, pend = pbeg + 4096;
    for (int p0 = pbeg; p0 < pend; p0 += 64) {   // 64-pixel chunk, single image row
        __builtin_prefetch(Gb + p0 + 64, 0, 0);  // -> global_prefetch_b8
        const int y  = p0 >> 8;
        const int xb = p0 & 255;
        // cooperative fill of T rows 0..7 (512 stencil/copy evals, 16 per lane)
        #pragma unroll
        for (int j = 0; j < 16; ++j) {
            int v = lane + 32 * j;               // 0..511
            int l = v >> 6, px = v & 63;
            float val;
            if (l == 7) {
                val = Gb[p0 + px];               // r row
            } else {
                float a = 0.f;
                int xx = xb + px;
                #pragma unroll
                for (int dy = -1; dy <= 1; ++dy) {
                    int y2 = y + dy;
                    if (y2 < 0 || y2 > 255) continue;
                    #pragma unroll
                    for (int dx = -1; dx <= 1; ++dx) {
                        int x2 = xx + dx;
                        if (x2 < 0 || x2 > 255) continue;
                        a += sk[(dy + 1) * 3 + (dx + 1)] * Gb[l * NPIX + (y2 << 8) + x2];
                    }
                }
                val = a;
            }
            T[l][px] = val;
        }
        __syncthreads();
        // T T^T: A-frag and B-frag have symmetric lane layouts -> same registers
        #pragma unroll
        for (int kc = 0; kc < 64; kc += 4) {
            v2f fr;
            fr.x = T[m][kc + khalf];
            fr.y = T[m][kc + khalf + 1];
            acc = __builtin_amdgcn_wmma_f32_16x16x4_f32(
                false, fr, false, fr, (short)0, acc, false, false);
        }
        __syncthreads();
    }
    // C layout: lanes 0..15 hold N=lane, M = vgpr index 0..7 (only 8x8 needed)
    if (lane < 8) {
        #pragma unroll
        for (int i = 0; i < 8; ++i)
            atomicAdd(&gram[b * 64 + i * 8 + lane], acc[i]);
    }
}

// ---------------- per-sample 7x7 Gauss-Jordan solve ------------------------
__global__ void solve_kernel(const float* __restrict__ gram, float* __restrict__ Kv) {
    const int b = blockIdx.x * blockDim.x + threadIdx.x;
    if (b >= B) return;
    float M[7][8];
    for (int i = 0; i < 7; ++i) {
        for (int j = 0; j < 7; ++j) M[i][j] = gram[b * 64 + i * 8 + j];
        M[i][7] = gram[b * 64 + i * 8 + 7];     // (S r)_i
    }
    for (int c = 0; c < 7; ++c) {
        int piv = c; float best = fabsf(M[c][c]);
        for (int rr = c + 1; rr < 7; ++rr) {
            float v = fabsf(M[rr][c]);
            if (v > best) { best = v; piv = rr; }
        }
        if (piv != c)
            for (int j = 0; j < 8; ++j) { float t = M[c][j]; M[c][j] = M[piv][j]; M[piv][j] = t; }
        float inv = 1.f / M[c][c];
        for (int j = c; j < 8; ++j) M[c][j] *= inv;
        for (int rr = 0; rr < 7; ++rr) {
            if (rr == c) continue;
            float fct = M[rr][c];
            for (int j = c; j < 8; ++j) M[rr][j] -= fct * M[c][j];
        }
    }
    for (int i = 0; i < 7; ++i) Kv[b * 8 + i] = M[i][7];
}

// ---------------- x_new = x + K^T G ----------------------------------------
__global__ void update_kernel(const float* __restrict__ x, const float* __restrict__ G,
                              const float* __restrict__ Kv, float* __restrict__ out) {
    const int b = blockIdx.y;
    const int p = blockIdx.x * 256 + threadIdx.x;
    __shared__ float k[7];
    if (threadIdx.x < 7) k[threadIdx.x] = Kv[b * 8 + threadIdx.x];
    __syncthreads();
    const float* Gb = G + (size_t)b * LCH * NPIX;
    float acc = x[(size_t)b * NPIX + p];
    #pragma unroll
    for (int l = 0; l < LCH; ++l) acc += k[l] * Gb[l * NPIX + p];
    out[(size_t)b * NPIX + p] = acc;
}

// ---------------------------------------------------------------------------
extern "C" void kernel_launch(void* const* d_in, const int* in_sizes, int n_in,
                              void* d_out, int out_size, void* d_ws, size_t ws_size,
                              hipStream_t stream) {
    const float* x      = (const float*)d_in[0];
    const float* f      = (const float*)d_in[1];
    const float* kA     = (const float*)d_in[2];
    const float* fc1_w1 = (const float*)d_in[3];
    const float* fc1_b1 = (const float*)d_in[4];
    const float* fc1_w2 = (const float*)d_in[5];
    const float* fc1_b2 = (const float*)d_in[6];
    const float* fc2_w1 = (const float*)d_in[7];
    const float* fc2_b1 = (const float*)d_in[8];
    const float* fc2_w2 = (const float*)d_in[9];
    const float* fc2_b2 = (const float*)d_in[10];
    float* out = (float*)d_out;

    float* W    = (float*)d_ws;
    float* G    = W;                         // 29,360,128 floats
    float* W1   = G + (size_t)B * LCH * NPIX;
    float* W2   = W1 + B * 147;
    float* gram = W2 + B * 441;
    float* Kv   = gram + B * 64;

    mlp_kernel<<<B, 256, 0, stream>>>(kA, fc1_w1, fc1_b1, fc1_w2, fc1_b2,
                                      fc2_w1, fc2_b1, fc2_w2, fc2_b2, W1, W2);
    residual_kernel<<<dim3(NPIX / 256, B), 256, 0, stream>>>(x, f, kA, G);
    conv1_kernel<<<dim3(MU / TS, MU / TS, B), 256, 0, stream>>>(W1, G);
    conv2_kernel<<<dim3(MU / TS, MU / TS, B), 256, 0, stream>>>(W2, G);
    zero_kernel<<<(B * 64 + 255) / 256, 256, 0, stream>>>(gram, B * 64);
    gram_kernel<<<dim3(16, B), 32, 0, stream>>>(kA, G, gram);
    solve_kernel<<<1, 64, 0, stream>>>(gram, Kv);
    update_kernel<<<dim3(NPIX / 256, B), 256, 0, stream>>>(x, G, Kv, out);
}